// GCN_87729001988319
// MI455X (gfx1250) — compile-verified
//
#include <hip/hip_runtime.h>

typedef __attribute__((ext_vector_type(2))) float v2f;
typedef __attribute__((ext_vector_type(8))) float v8f;

#define D_IN  128
#define D_HID 128
#define D_EMB 64
#define D_OUT 2

// ---------------------------------------------------------------- utilities
__global__ void init_deg(float* __restrict__ deg, int n) {
  int i = blockIdx.x * blockDim.x + threadIdx.x;
  if (i < n) deg[i] = 1.0f;                       // self-loop contributes 1
}

__global__ void deg_edges(const int* __restrict__ dst, float* __restrict__ deg, int e) {
  int i = blockIdx.x * blockDim.x + threadIdx.x;
  if (i < e) atomicAdd(&deg[dst[i]], 1.0f);
}

__global__ void make_dinv(float* __restrict__ deg, int n) {
  int i = blockIdx.x * blockDim.x + threadIdx.x;
  if (i < n) deg[i] = rsqrtf(deg[i]);             // deg >= 1 always (self-loops)
}

__global__ void make_norm(const int* __restrict__ src, const int* __restrict__ dst,
                          const float* __restrict__ dinv, float* __restrict__ norm,
                          int e, int n) {
  int i = blockIdx.x * blockDim.x + threadIdx.x;
  if (i < e) {
    norm[i] = dinv[src[i]] * dinv[dst[i]];
  } else if (i < e + n) {
    float d = dinv[i - e];
    norm[i] = d * d;                              // self-loop edge
  }
}

// Initialize accumulator with broadcast bias (fuses the zero + bias passes).
template <int D>
__global__ void init_bias(float* __restrict__ H, const float* __restrict__ b,
                          long long total) {
  long long i = (long long)blockIdx.x * blockDim.x + threadIdx.x;
  if (i < total) H[i] = b[(int)(i & (D - 1))];
}

// ---------------------------------------------------------- weight repacking
// Wp[(h*DO + n)*(K/2) + (k0/4)*2 + j] = W[(k0 + 2h + j)*DO + n]
// -> each WMMA lane's B fragments across the K loop become contiguous, so the
//    GEMM loads one float4 (global_load_b128) per TWO v_wmma instead of four
//    512B-strided global_load_b32.
template <int K, int DO>
__global__ void repack_W(const float* __restrict__ W, float* __restrict__ Wp) {
  constexpr int HALFSZ = DO * (K / 2);
  int tid = blockIdx.x * blockDim.x + threadIdx.x;
  if (tid >= 2 * HALFSZ) return;
  int h   = tid / HALFSZ;
  int rem = tid - h * HALFSZ;
  int n   = rem / (K / 2);
  int q   = rem - n * (K / 2);
  int k   = (q >> 1) * 4 + 2 * h + (q & 1);
  Wp[tid] = W[(size_t)k * DO + n];
}

// ------------------------------------------------- fp32 WMMA GEMM: Out = X @ Wp
// One wave computes a 16x16 tile of Out. blockIdx.x = M-tile, wave id = N-tile.
// The 16xK A tile is shared by all waves -> staged once in LDS (padded stride
// K+4 floats: lane r reads banks {4r+kh, 4r+kh+1} -> all 64 banks, no conflict;
// backend fuses pairs into ds_load_2addr_b64). B comes from the repacked Wp as
// one b128 per two WMMAs. RELU_A applies ReLU to A on load (previous layer's
// activation folded in).
// A (16x4 f32):  lanes 0-15 -> M=lane, K={0,1}; lanes 16-31 -> M=lane-16, K={2,3}
// B (4x16 f32):  lanes 0-15 -> N=lane, K={0,1}; lanes 16-31 -> N=lane-16, K={2,3}
// C/D (16x16):   VGPR r: lanes 0-15 -> (M=r, N=lane); lanes 16-31 -> (M=r+8, N=lane-16)
template <int K, int DO, bool RELU_A>
__global__ __launch_bounds__(32 * (DO / 16))
void gemm_wmma_f32(const float* __restrict__ X, const float* __restrict__ Wp,
                   float* __restrict__ Out) {
  constexpr int NT  = 32 * (DO / 16);             // threads per block
  constexpr int LDA = K + 4;                      // padded LDS row stride (floats)
  __shared__ float sA[16 * LDA];

  const int mbase = blockIdx.x * 16;

  // cooperative A-tile fill, float2 granularity (8B aligned: LDA even)
  for (int c = threadIdx.x; c < 16 * (K / 2); c += NT) {
    int row  = c / (K / 2);
    int col2 = c % (K / 2);
    float2 v = *(const float2*)(X + (size_t)(mbase + row) * K + col2 * 2);
    if (RELU_A) { v.x = fmaxf(v.x, 0.0f); v.y = fmaxf(v.y, 0.0f); }
    *(float2*)(sA + row * LDA + col2 * 2) = v;
  }
  __syncthreads();

  const int lane = threadIdx.x & 31;
  const int wave = threadIdx.x >> 5;              // N-tile index
  const int half = lane >> 4;                     // 0 or 1 (upper K pair)
  const int l15  = lane & 15;
  const int n    = wave * 16 + l15;

  const float* arow = sA + l15 * LDA + half * 2;
  const float* bcol = Wp + (size_t)(half * DO + n) * (K / 2);  // contiguous frags

  v8f c = {};
#pragma unroll
  for (int k0 = 0; k0 < K; k0 += 8) {
    float4 bq = *(const float4*)(bcol + (k0 >> 1));   // b128, 16B aligned
    v2f a0 = *(const v2f*)(arow + k0);                // ds b64, conflict-free
    v2f a1 = *(const v2f*)(arow + k0 + 4);
    v2f b0; b0.x = bq.x; b0.y = bq.y;
    v2f b1; b1.x = bq.z; b1.y = bq.w;
    c = __builtin_amdgcn_wmma_f32_16x16x4_f32(false, a0, false, b0,
                                              (short)0, c, false, false);
    c = __builtin_amdgcn_wmma_f32_16x16x4_f32(false, a1, false, b1,
                                              (short)0, c, false, false);
  }
  const int rbase = mbase + half * 8;
#pragma unroll
  for (int r = 0; r < 8; ++r)
    Out[(size_t)(rbase + r) * DO + n] = c[r];
}

// ---------------------------------------- edge scatter: Agg[dst] += T[src]*norm
// D/4 threads per edge, each thread handles 4 contiguous features (float4 gather,
// 4x global_atomic_add_f32). T and Agg (<= 20.5 MB) are L2-resident (192 MB L2).
template <int D>
__global__ void scatter_edges(const float* __restrict__ T, const int* __restrict__ src,
                              const int* __restrict__ dst, const float* __restrict__ norm,
                              float* __restrict__ Agg, int e_total, int e) {
  constexpr int CPE = D / 4;                      // chunks per edge (power of 2)
  long long t = (long long)blockIdx.x * blockDim.x + threadIdx.x;
  int edge = (int)(t / CPE);
  int ch   = (int)(t % CPE);
  if (edge >= e_total) return;
  int s, d;
  if (edge < e) { s = src[edge]; d = dst[edge]; }
  else          { s = d = edge - e; }             // self-loop
  float w = norm[edge];
  const float4 v = *(const float4*)(T + (size_t)s * D + ch * 4);
  float* out = Agg + (size_t)d * D + ch * 4;
  atomicAdd(out + 0, v.x * w);
  atomicAdd(out + 1, v.y * w);
  atomicAdd(out + 2, v.z * w);
  atomicAdd(out + 3, v.w * w);
}

// ------------------------------------------------ output projection [N,64]x[64,2]
__global__ void out_proj(const float* __restrict__ H, const float* __restrict__ Wo,
                         const float* __restrict__ bo, float* __restrict__ Out, int n) {
  int i = blockIdx.x * blockDim.x + threadIdx.x;
  if (i >= n) return;
  const float* h = H + (size_t)i * D_EMB;
  float a0 = bo[0], a1 = bo[1];
#pragma unroll
  for (int k4 = 0; k4 < D_EMB; k4 += 4) {
    float4 hv = *(const float4*)(h + k4);
    a0 = fmaf(hv.x, Wo[(k4 + 0) * 2 + 0], a0);
    a1 = fmaf(hv.x, Wo[(k4 + 0) * 2 + 1], a1);
    a0 = fmaf(hv.y, Wo[(k4 + 1) * 2 + 0], a0);
    a1 = fmaf(hv.y, Wo[(k4 + 1) * 2 + 1], a1);
    a0 = fmaf(hv.z, Wo[(k4 + 2) * 2 + 0], a0);
    a1 = fmaf(hv.z, Wo[(k4 + 2) * 2 + 1], a1);
    a0 = fmaf(hv.w, Wo[(k4 + 3) * 2 + 0], a0);
    a1 = fmaf(hv.w, Wo[(k4 + 3) * 2 + 1], a1);
  }
  Out[i * 2 + 0] = a0;
  Out[i * 2 + 1] = a1;
}

// ---------------------------------------------------------------------------
static inline int cdiv(long long a, long long b) { return (int)((a + b - 1) / b); }

extern "C" void kernel_launch(void* const* d_in, const int* in_sizes, int n_in,
                              void* d_out, int out_size, void* d_ws, size_t ws_size,
                              hipStream_t stream) {
  const float* x  = (const float*)d_in[0];
  const int*   ei = (const int*)d_in[1];          // [2, E] flat: row0=src, row1=dst
  const float* W1 = (const float*)d_in[2];
  const float* b1 = (const float*)d_in[3];
  const float* W2 = (const float*)d_in[4];
  const float* b2 = (const float*)d_in[5];
  const float* W3 = (const float*)d_in[6];
  const float* b3 = (const float*)d_in[7];
  const float* Wo = (const float*)d_in[8];
  const float* bo = (const float*)d_in[9];

  const int N  = in_sizes[0] / D_IN;
  const int E  = in_sizes[1] / 2;
  const int ET = E + N;                           // edges incl. self-loops
  const int* src = ei;
  const int* dst = ei + E;

  // workspace (floats): dinv[N] | norm[ET] | T[N*128] | H[N*128] | Wp1 | Wp2 | Wp3
  float* wsp  = (float*)d_ws;
  float* dinv = wsp;
  float* norm = dinv + N;
  float* T    = norm + ET;
  float* H    = T + (size_t)N * D_HID;
  float* Wp1  = H + (size_t)N * D_HID;
  float* Wp2  = Wp1 + D_IN * D_HID;
  float* Wp3  = Wp2 + D_HID * D_HID;

  float* outp = (float*)d_out;                    // [N,2]
  float* h3   = outp + (size_t)N * D_OUT;         // [N,64]  (second tuple element)

  // ---- weight repack into per-lane WMMA fragment order (160 KB total)
  repack_W<D_IN,  D_HID><<<cdiv(D_IN * D_HID, 256), 256, 0, stream>>>(W1, Wp1);
  repack_W<D_HID, D_HID><<<cdiv(D_HID * D_HID, 256), 256, 0, stream>>>(W2, Wp2);
  repack_W<D_HID, D_EMB><<<cdiv(D_HID * D_EMB, 256), 256, 0, stream>>>(W3, Wp3);

  // ---- normalization coefficients
  init_deg<<<cdiv(N, 256), 256, 0, stream>>>(dinv, N);
  deg_edges<<<cdiv(E, 256), 256, 0, stream>>>(dst, dinv, E);
  make_dinv<<<cdiv(N, 256), 256, 0, stream>>>(dinv, N);
  make_norm<<<cdiv(ET, 256), 256, 0, stream>>>(src, dst, dinv, norm, E, N);

  const long long nh = (long long)N * D_HID;
  const long long ne = (long long)N * D_EMB;

  // ---- layer 1: T = x@W1 ; H = b1 + scatter(T)   (ReLU deferred to next GEMM)
  gemm_wmma_f32<D_IN, D_HID, false><<<N / 16, 256, 0, stream>>>(x, Wp1, T);
  init_bias<D_HID><<<cdiv(nh, 256), 256, 0, stream>>>(H, b1, nh);
  scatter_edges<D_HID><<<cdiv((long long)ET * (D_HID / 4), 256), 256, 0, stream>>>(
      T, src, dst, norm, H, ET, E);

  // ---- layer 2: T = relu(H)@W2 ; H = b2 + scatter(T)
  gemm_wmma_f32<D_HID, D_HID, true><<<N / 16, 256, 0, stream>>>(H, Wp2, T);
  init_bias<D_HID><<<cdiv(nh, 256), 256, 0, stream>>>(H, b2, nh);
  scatter_edges<D_HID><<<cdiv((long long)ET * (D_HID / 4), 256), 256, 0, stream>>>(
      T, src, dst, norm, H, ET, E);

  // ---- layer 3: T = relu(H)@W3 ; h3 = b3 + scatter(T)  (no relu), into d_out
  gemm_wmma_f32<D_HID, D_EMB, true><<<N / 16, 128, 0, stream>>>(H, Wp3, T);
  init_bias<D_EMB><<<cdiv(ne, 256), 256, 0, stream>>>(h3, b3, ne);
  scatter_edges<D_EMB><<<cdiv((long long)ET * (D_EMB / 4), 256), 256, 0, stream>>>(
      T, src, dst, norm, h3, ET, E);

  // ---- output projection: out = h3 @ Wout + bout
  out_proj<<<cdiv(N, 256), 256, 0, stream>>>(h3, Wo, bo, outp, N);
}